// AdvancedLearnableEntropyPooling2D_11965778887255
// MI455X (gfx1250) — compile-verified
//
#include <hip/hip_runtime.h>

// Problem shape (fixed by the reference harness)
constexpr int Bn = 32;
constexpr int Hn = 256;
constexpr int Wn = 256;
constexpr int Cn = 128;
constexpr int HOn = 128;   // (256-2)/2+1
constexpr int WOn = 128;
constexpr int TW  = 8;            // output columns per block
constexpr int IN_COLS = 2 * TW;   // 16 input columns staged per block
constexpr float EPSv = 1e-6f;

// Tile: 2 input rows x 16 input cols x 128 channels = 4096 floats = 16 KB.
// Staged global->LDS via the CDNA5 async copy path (ASYNCcnt), marked
// non-temporal: every input byte is touched exactly once (2x2 stride-2
// windows are non-overlapping), so the 1 GiB stream should not occupy L2.
__global__ __launch_bounds__(256) void
entropy_pool2d_kernel(const float* __restrict__ x,
                      const float* __restrict__ ew,
                      const float* __restrict__ pb,
                      float* __restrict__ out)
{
    __shared__ float tile[2 * IN_COLS * Cn];   // [row(2)][col(16)][c(128)]

    const int t   = threadIdx.x;               // 0..255
    const unsigned blk = blockIdx.x;           // 0..65535
    const int wt  = blk & 15;                  // tile index along Wo (Wo/TW = 16)
    const int ho  = (blk >> 4) & (HOn - 1);
    const int b   = blk >> 11;

    const int h0  = ho * 2;                    // first input row
    const int w0  = wt * IN_COLS;              // first input column

    // LDS byte offset of `tile`: generic shared pointers live in the LDS
    // aperture, whose low 32 bits are the LDS byte offset.
    const unsigned lds_base = (unsigned)(size_t)(void*)tile;

    // ---- async stage: 2 rows, each 16 cols x 128 ch = 2048 floats (512 x b128)
    // 256 threads -> 2 b128 per thread per row, fully coalesced 16B/lane.
#pragma unroll
    for (int r = 0; r < 2; ++r) {
        const float* grow = x + ((size_t)((b * Hn + h0 + r) * Wn + w0) * Cn);
#pragma unroll
        for (int i = 0; i < 2; ++i) {
            const int j = t + i * 256;                       // float4 index [0,512)
            const unsigned loff = lds_base + (unsigned)((r * (IN_COLS * Cn) + j * 4) * 4);
            const float* gaddr = grow + (size_t)j * 4;       // 16B aligned
            asm volatile("global_load_async_to_lds_b128 %0, %1, off th:TH_LOAD_NT"
                         :: "v"(loff), "v"(gaddr)
                         : "memory");
        }
    }
#if __has_builtin(__builtin_amdgcn_s_wait_asynccnt)
    __builtin_amdgcn_s_wait_asynccnt(0);
#else
    asm volatile("s_wait_asynccnt 0" ::: "memory");
#endif
    __syncthreads();

    // ---- compute: per thread, fixed channel; 4 output columns
    const int c   = t & (Cn - 1);
    const int wl0 = t >> 7;                    // 0 or 1
    const float wgt = ew[c];
    const float bia = pb[c];

#pragma unroll
    for (int k = 0; k < 4; ++k) {
        const int wl  = wl0 + 2 * k;           // local output col 0..7
        const int col = 2 * wl;                // local input col
        const float x0 = tile[(0 * IN_COLS + col    ) * Cn + c];
        const float x1 = tile[(0 * IN_COLS + col + 1) * Cn + c];
        const float x2 = tile[(1 * IN_COLS + col    ) * Cn + c];
        const float x3 = tile[(1 * IN_COLS + col + 1) * Cn + c];

        // stabilized softmax over the 4 window elements
        const float m  = fmaxf(fmaxf(x0, x1), fmaxf(x2, x3));
        const float e0 = __expf(x0 - m);
        const float e1 = __expf(x1 - m);
        const float e2 = __expf(x2 - m);
        const float e3 = __expf(x3 - m);
        const float inv = 1.0f / (e0 + e1 + e2 + e3);
        const float p0 = e0 * inv, p1 = e1 * inv, p2 = e2 * inv, p3 = e3 * inv;

        // entropy = -sum p*log(p + EPS)  (matches reference)
        const float ent = -(p0 * __logf(p0 + EPSv) + p1 * __logf(p1 + EPSv) +
                            p2 * __logf(p2 + EPSv) + p3 * __logf(p3 + EPSv));

        const size_t oidx = ((size_t)((b * HOn + ho) * WOn) + (wt * TW + wl)) * Cn + c;
        __builtin_nontemporal_store(ent * wgt + bia, &out[oidx]);
    }
}

extern "C" void kernel_launch(void* const* d_in, const int* in_sizes, int n_in,
                              void* d_out, int out_size, void* d_ws, size_t ws_size,
                              hipStream_t stream)
{
    (void)in_sizes; (void)n_in; (void)d_ws; (void)ws_size; (void)out_size;
    const float* x  = (const float*)d_in[0];
    const float* ew = (const float*)d_in[1];
    const float* pb = (const float*)d_in[2];
    float* out = (float*)d_out;

    // 32 * 128 * (128/8) = 65536 blocks, 256 threads (8 wave32) each.
    const int nblocks = Bn * HOn * (WOn / TW);
    entropy_pool2d_kernel<<<nblocks, 256, 0, stream>>>(x, ew, pb, out);
}